// QuantumPatchwise_Angle_91173565759919
// MI455X (gfx1250) — compile-verified
//
#include <hip/hip_runtime.h>
#include <hip/hip_bf16.h>

// ---------------- problem constants ----------------
#define N_QUBITS 12
#define NSTATE   4096          // 2^12
#define NPATCH   196           // (224/16)^2
#define BATCH    32
#define NSAMP    (BATCH * NPATCH)      // 6272
#define NCLS     1000
#define KCLS     (NPATCH * N_QUBITS)   // 2352
#define IMG      224
#define PATCH    16

typedef __attribute__((ext_vector_type(2))) float v2f;
typedef __attribute__((ext_vector_type(8))) float v8f;

// Device-pass-only sanity check (round-1 feedback showed the builtin exists on
// the gfx1250 device pass; the host pass must never evaluate it).
#if defined(__HIP_DEVICE_COMPILE__) && \
    !__has_builtin(__builtin_amdgcn_wmma_f32_16x16x4_f32)
#error "missing __builtin_amdgcn_wmma_f32_16x16x4_f32 for gfx1250"
#endif

__device__ __forceinline__ v8f wmma4(v2f a, v2f b, v8f c) {
#if __has_builtin(__builtin_amdgcn_wmma_f32_16x16x4_f32)
  // D = A(16x4) * B(4x16) + C ; f32 WMMA, 8-arg form
  return __builtin_amdgcn_wmma_f32_16x16x4_f32(false, a, false, b, (short)0, c,
                                               false, false);
#else
  (void)a; (void)b;
  return c;  // host pass stub: never executed on device
#endif
}

// ---------------- fused-op descriptors (7 ops per circuit layer) ------------
// Gate application order inside an op follows the reference gate list order.
// local bit 3 (MSB of the 4-bit z index) = smallest qubit number = highest
// global state bit (global bit of qubit q is 11-q).
__constant__ int OP_NG[7]    = {4, 4, 4, 2, 2, 2, 2};
__constant__ int OP_GP[7][4] = {{0, 1, 6, 7},   {2, 3, 8, 9},  {4, 5, 10, 11},
                                {12, 13, 0, 0}, {14, 15, 0, 0},
                                {16, 17, 0, 0}, {18, 19, 0, 0}};
__constant__ int OP_L1[7][4] = {{3, 1, 3, 2}, {3, 1, 3, 2}, {3, 1, 3, 2},
                                {3, 2, 0, 0}, {3, 2, 0, 0},
                                {3, 2, 0, 0}, {3, 2, 0, 0}};
__constant__ int OP_L2[7][4] = {{2, 0, 1, 0}, {2, 0, 1, 0}, {2, 0, 1, 0},
                                {1, 0, 0, 0}, {1, 0, 0, 0},
                                {1, 0, 0, 0}, {1, 0, 0, 0}};
__constant__ int OP_ZP[7][4] = {{11, 10, 9, 8}, {7, 6, 5, 4}, {3, 2, 1, 0},
                                {11, 10, 7, 6}, {9, 8, 5, 4},
                                {11, 10, 3, 2}, {9, 8, 1, 0}};

// ---------------- kernel 0: build the 14 fused 16x16 unitaries --------------
// RBS-derived 4x4 gate rows (U = CNOT * kron(ry_m,I) * CNOT * kron(ry_p,I)):
//   row0: [ c^2,  s^2, -cs,  cs]
//   row1: [ s^2,  c^2,  cs, -cs]
//   row2: [ cs , -cs ,  c^2, s^2]
//   row3: [-cs ,  cs ,  s^2, c^2]
__global__ void build_w_kernel(const float* __restrict__ qp,
                               float* __restrict__ Wg) {
  int op = blockIdx.x;            // 0..13
  int layer = op / 7, o = op % 7;
  int t = threadIdx.x;            // 0..255
  int x = t >> 4, y = t & 15;
  __shared__ float Wb[2][256];
  Wb[0][t] = (x == y) ? 1.0f : 0.0f;
  __syncthreads();
  int cur = 0;
  int ng = OP_NG[o];
  for (int g = 0; g < ng; ++g) {
    float th = qp[layer * 32 + OP_GP[o][g]];
    float c = cosf(th), s = sinf(th);
    float cc = c * c, ss = s * s, cs = c * s;
    int l1 = OP_L1[o][g], l2 = OP_L2[o][g];
    int xb = (((x >> l1) & 1) << 1) | ((x >> l2) & 1);
    float u0, u1, u2, u3;
    if (xb == 0)      { u0 =  cc; u1 =  ss; u2 = -cs; u3 =  cs; }
    else if (xb == 1) { u0 =  ss; u1 =  cc; u2 =  cs; u3 = -cs; }
    else if (xb == 2) { u0 =  cs; u1 = -cs; u2 =  cc; u3 =  ss; }
    else              { u0 = -cs; u1 =  cs; u2 =  ss; u3 =  cc; }
    int m1 = 1 << l1, m2 = 1 << l2;
    int bx = x & ~(m1 | m2);
    float acc = u0 * Wb[cur][bx * 16 + y] +
                u1 * Wb[cur][(bx | m2) * 16 + y] +
                u2 * Wb[cur][(bx | m1) * 16 + y] +
                u3 * Wb[cur][(bx | m1 | m2) * 16 + y];
    Wb[1 - cur][t] = acc;
    __syncthreads();
    cur ^= 1;
  }
  Wg[op * 256 + t] = Wb[cur][t];
}

// ---------------- kernel 1: patch mean + projection -> (cos,sin) amps -------
__global__ void angles_kernel(const float* __restrict__ x,
                              const float* __restrict__ pw,
                              const float* __restrict__ pb,
                              float* __restrict__ amps) {
  int gid = blockIdx.x * blockDim.x + threadIdx.x;
  if (gid >= NSAMP * N_QUBITS) return;
  int q = gid % N_QUBITS;
  int s = gid / N_QUBITS;
  int b = s / NPATCH, p = s % NPATCH;
  int ph = p / 14, pwc = p % 14;
  const float* xb = x + (size_t)b * 3 * IMG * IMG;
  const float* wr = pw + q * (PATCH * PATCH);
  float acc = 0.0f;
  for (int i = 0; i < PATCH; ++i) {
    const float* x0 = xb + (ph * PATCH + i) * IMG + pwc * PATCH;
    for (int j = 0; j < PATCH; ++j) {
      float g = (x0[j] + x0[j + IMG * IMG] + x0[j + 2 * IMG * IMG]) *
                (1.0f / 3.0f);
      acc += g * wr[i * PATCH + j];
    }
  }
  float h = 0.5f * (acc + pb[q]);
  amps[s * 24 + q * 2 + 0] = cosf(h);
  amps[s * 24 + q * 2 + 1] = sinf(h);
}

// ---------------- kernel 2: quantum state evolution in LDS via f32 WMMA -----
__global__ void __launch_bounds__(256)
quantum_kernel(const float* __restrict__ amps, const float* __restrict__ Wg,
               float* __restrict__ feats) {
  int s = blockIdx.x;
  int t = threadIdx.x;
  int lane = t & 31, wave = t >> 5;
  __shared__ float st[NSTATE];     // 16 KB state, later reused as 12x256 scratch
  __shared__ int zoffS[112];       // per-op row-scatter tables (7 ops x 16)
  __shared__ float ampsS[24];

  if (t < 24) ampsS[t] = amps[s * 24 + t];
  if (t < 112) {
    int o = t >> 4, z = t & 15;
    zoffS[t] = (((z >> 3) & 1) << OP_ZP[o][0]) |
               (((z >> 2) & 1) << OP_ZP[o][1]) |
               (((z >> 1) & 1) << OP_ZP[o][2]) |
               ((z & 1) << OP_ZP[o][3]);
  }
  __syncthreads();

  // initial product state: idx = t*16 + k ; qubit q lives at bit 11-q
  {
    float phi = 1.0f;
    for (int q = 0; q < 8; ++q) phi *= ampsS[2 * q + ((t >> (7 - q)) & 1)];
    for (int k = 0; k < 16; ++k) {
      st[t * 16 + k] = phi * ampsS[16 + ((k >> 3) & 1)] *
                       ampsS[18 + ((k >> 2) & 1)] *
                       ampsS[20 + ((k >> 1) & 1)] * ampsS[22 + (k & 1)];
    }
  }
  __syncthreads();

  int koff = (lane < 16) ? 0 : 2;   // A/B K-half per documented f32 layout
  int lrow = lane & 15;

  for (int opi = 0; opi < 14; ++opi) {
    int o = opi % 7;
    const float* Wop = Wg + opi * 256;
    // A fragments: A_k = W[:, 4k..4k+3]; v0 holds K%4 in {0,2}, v1 in {1,3}
    v2f a0, a1, a2, a3;
    a0.x = Wop[lrow * 16 + koff + 0];  a0.y = Wop[lrow * 16 + koff + 1];
    a1.x = Wop[lrow * 16 + koff + 4];  a1.y = Wop[lrow * 16 + koff + 5];
    a2.x = Wop[lrow * 16 + koff + 8];  a2.y = Wop[lrow * 16 + koff + 9];
    a3.x = Wop[lrow * 16 + koff + 12]; a3.y = Wop[lrow * 16 + koff + 13];
    int zmask = (1 << OP_ZP[o][0]) | (1 << OP_ZP[o][1]) |
                (1 << OP_ZP[o][2]) | (1 << OP_ZP[o][3]);
    const int* zt = zoffS + o * 16;
    for (int tt = 0; tt < 2; ++tt) {
      int col = (wave + 8 * tt) * 16 + lrow;   // this lane's state column
      // scatter the 8 column bits over the non-z bit positions
      int coff = 0, rb = 7;
      for (int pbit = 11; pbit >= 0; --pbit) {
        if (!((zmask >> pbit) & 1)) {
          coff |= ((col >> rb) & 1) << pbit;
          --rb;
        }
      }
      // B fragments: rows 4k+koff, 4k+1+koff of S for this column
      v2f b0, b1, b2, b3;
      b0.x = st[coff + zt[0 + koff]];  b0.y = st[coff + zt[1 + koff]];
      b1.x = st[coff + zt[4 + koff]];  b1.y = st[coff + zt[5 + koff]];
      b2.x = st[coff + zt[8 + koff]];  b2.y = st[coff + zt[9 + koff]];
      b3.x = st[coff + zt[12 + koff]]; b3.y = st[coff + zt[13 + koff]];
      v8f acc = {};
      acc = wmma4(a0, b0, acc);
      acc = wmma4(a1, b1, acc);
      acc = wmma4(a2, b2, acc);
      acc = wmma4(a3, b3, acc);
      // D: VGPR v -> row v (lanes<16) / row v+8 (lanes>=16), same column
      int mo = (lane < 16) ? 0 : 8;
      st[coff + zt[mo + 0]] = acc[0];
      st[coff + zt[mo + 1]] = acc[1];
      st[coff + zt[mo + 2]] = acc[2];
      st[coff + zt[mo + 3]] = acc[3];
      st[coff + zt[mo + 4]] = acc[4];
      st[coff + zt[mo + 5]] = acc[5];
      st[coff + zt[mo + 6]] = acc[6];
      st[coff + zt[mo + 7]] = acc[7];
    }
    __syncthreads();
  }

  // Z expectations: z_q = sum_{bit=0} p - sum_{bit=1} p
  float pk[16];
  for (int k = 0; k < 16; ++k) {
    float v = st[t * 16 + k];
    pk[k] = v * v;
  }
  float S = 0.0f;
  for (int k = 0; k < 16; ++k) S += pk[k];
  float z8 = 0, z9 = 0, z10 = 0, z11 = 0;
  for (int k = 0; k < 16; ++k) {
    z8  += ((k >> 3) & 1) ? -pk[k] : pk[k];
    z9  += ((k >> 2) & 1) ? -pk[k] : pk[k];
    z10 += ((k >> 1) & 1) ? -pk[k] : pk[k];
    z11 += (k & 1)        ? -pk[k] : pk[k];
  }
  __syncthreads();                 // all reads of state done; reuse as scratch
  for (int q = 0; q < 8; ++q) {
    float sgn = ((t >> (7 - q)) & 1) ? -1.0f : 1.0f;
    st[q * 256 + t] = sgn * S;
  }
  st[8 * 256 + t] = z8;
  st[9 * 256 + t] = z9;
  st[10 * 256 + t] = z10;
  st[11 * 256 + t] = z11;
  __syncthreads();
  for (int sr = 128; sr > 0; sr >>= 1) {
    if (t < sr)
      for (int q = 0; q < N_QUBITS; ++q)
        st[q * 256 + t] += st[q * 256 + t + sr];
    __syncthreads();
  }
  if (t < N_QUBITS) feats[s * N_QUBITS + t] = st[t * 256];
}

// ---------------- kernel 3: classifier GEMM (32x2352 @ 2352x1000) via WMMA --
__global__ void __launch_bounds__(32)
cls_kernel(const float* __restrict__ feats, const float* __restrict__ w,
           const float* __restrict__ bias, float* __restrict__ out) {
  int lane = threadIdx.x;
  int mb = blockIdx.x * 16;
  int nb = blockIdx.y * 16;
  int koff = (lane < 16) ? 0 : 2;
  int lrow = lane & 15;
  const float* ap = feats + (size_t)(mb + lrow) * KCLS;   // A row m
  int bcol = nb + lrow;
  int bcolc = (bcol < NCLS) ? bcol : (NCLS - 1);
  const float* bp = w + (size_t)bcolc * KCLS;             // B col n = cls_w row
  v8f acc = {};
  for (int k = 0; k < KCLS; k += 4) {
    v2f a; a.x = ap[k + koff]; a.y = ap[k + koff + 1];
    v2f b; b.x = bp[k + koff]; b.y = bp[k + koff + 1];
    acc = wmma4(a, b, acc);
  }
  int n = nb + lrow;
  int m0 = mb + ((lane < 16) ? 0 : 8);
  if (n < NCLS) {
    float bv = bias[n];
    out[(m0 + 0) * NCLS + n] = acc[0] + bv;
    out[(m0 + 1) * NCLS + n] = acc[1] + bv;
    out[(m0 + 2) * NCLS + n] = acc[2] + bv;
    out[(m0 + 3) * NCLS + n] = acc[3] + bv;
    out[(m0 + 4) * NCLS + n] = acc[4] + bv;
    out[(m0 + 5) * NCLS + n] = acc[5] + bv;
    out[(m0 + 6) * NCLS + n] = acc[6] + bv;
    out[(m0 + 7) * NCLS + n] = acc[7] + bv;
  }
}

// ---------------- launcher ----------------
extern "C" void kernel_launch(void* const* d_in, const int* in_sizes, int n_in,
                              void* d_out, int out_size, void* d_ws,
                              size_t ws_size, hipStream_t stream) {
  (void)in_sizes; (void)n_in; (void)out_size; (void)ws_size;
  const float* x  = (const float*)d_in[0];
  const float* pw = (const float*)d_in[1];
  const float* pb = (const float*)d_in[2];
  const float* qp = (const float*)d_in[3];
  const float* cw = (const float*)d_in[4];
  const float* cb = (const float*)d_in[5];
  float* out = (float*)d_out;
  float* ws = (float*)d_ws;

  float* Wg    = ws;                   // 14*256 floats  (fused unitaries)
  float* amps  = ws + 4096;            // NSAMP*24 floats (cos/sin per qubit)
  float* feats = amps + NSAMP * 24;    // NSAMP*12 floats (Z expectations)

  build_w_kernel<<<14, 256, 0, stream>>>(qp, Wg);
  angles_kernel<<<(NSAMP * N_QUBITS + 255) / 256, 256, 0, stream>>>(x, pw, pb,
                                                                    amps);
  quantum_kernel<<<NSAMP, 256, 0, stream>>>(amps, Wg, feats);
  cls_kernel<<<dim3(BATCH / 16, (NCLS + 15) / 16), 32, 0, stream>>>(feats, cw,
                                                                    cb, out);
}